// SummaryNetLSTM_47914655154283
// MI455X (gfx1250) — compile-verified
//
#include <hip/hip_runtime.h>
#include <hip/hip_bf16.h>

// ---------------------------------------------------------------------------
// SummaryNetLSTM for MI455X (gfx1250, wave32)
//   B=65536 samples, T=100, C=7, H=2 (bi-LSTM x2) -> relu -> FC 400->64 -> 20
// LSTM part: one sample per lane, scalar FMA + v_exp_f32 / v_rcp_f32.
// FC part:   V_WMMA_F32_16X16X4_F32 (full-precision fp32 WMMA) GEMMs.
// ---------------------------------------------------------------------------

#define NB 65536   // batch
#define NT 100     // time steps

typedef __attribute__((ext_vector_type(2))) float v2f;
typedef __attribute__((ext_vector_type(8))) float v8f;

// Fast reciprocal via hardware v_rcp_f32 (trans unit, co-executes with FMA).
__device__ __forceinline__ float fast_rcp(float x) {
  return __builtin_amdgcn_rcpf(x);
}
__device__ __forceinline__ float sigm(float x) {
  return fast_rcp(1.0f + __expf(-x));
}
__device__ __forceinline__ float tanh_fast(float x) {
  // tanh(x) = 2/(1+exp(-2x)) - 1 ; exact at both saturation ends
  return fmaf(2.0f, fast_rcp(1.0f + __expf(-2.0f * x)), -1.0f);
}

// ---------------------------------------------------------------------------
// One LSTM direction. IN = input channels (7 for layer0, 4 for layer1).
// TMAJOR=false: input is (B, T, IN) row-major (the raw x tensor).
// TMAJOR=true : input is (T, B, 4) (our intermediate layout), IN must be 4.
// Output: (T, B, 4) floats, writing channel pair [out_off, out_off+1].
// Gate order per jnp.split: rows 0-1 = i, 2-3 = f, 4-5 = g, 6-7 = o.
// ---------------------------------------------------------------------------
template <int IN, bool REV, bool TMAJOR>
__global__ __launch_bounds__(256) void lstm_dir_kernel(
    const float* __restrict__ in,
    const float* __restrict__ w_ih,   // (8, IN)
    const float* __restrict__ w_hh,   // (8, 2)
    const float* __restrict__ b_ih,   // (8)
    const float* __restrict__ b_hh,   // (8)
    float* __restrict__ out,          // (T, B, 4)
    int out_off)
{
  const int b = blockIdx.x * blockDim.x + threadIdx.x;
  if (b >= NB) return;

  float wih[8][IN];
  float whh[8][2];
  float bias[8];
#pragma unroll
  for (int g = 0; g < 8; ++g) {
#pragma unroll
    for (int c = 0; c < IN; ++c) wih[g][c] = w_ih[g * IN + c];
    whh[g][0] = w_hh[g * 2 + 0];
    whh[g][1] = w_hh[g * 2 + 1];
    bias[g] = b_ih[g] + b_hh[g];
  }

  float h0 = 0.f, h1 = 0.f, c0 = 0.f, c1 = 0.f;

  for (int it = 0; it < NT; ++it) {
    const int tt = REV ? (NT - 1 - it) : it;

    float xv[IN];
    if constexpr (TMAJOR) {
      const float4 x4 =
          *(const float4*)(in + ((size_t)tt * NB + (size_t)b) * 4);
      xv[0] = x4.x; xv[1] = x4.y; xv[2] = x4.z; xv[3] = x4.w;
    } else {
      const float* p = in + (size_t)b * NT * IN + (size_t)tt * IN;
#pragma unroll
      for (int c = 0; c < IN; ++c) xv[c] = p[c];
    }

    float g[8];
#pragma unroll
    for (int gg = 0; gg < 8; ++gg) {
      float acc = fmaf(whh[gg][0], h0, fmaf(whh[gg][1], h1, bias[gg]));
#pragma unroll
      for (int c = 0; c < IN; ++c) acc = fmaf(wih[gg][c], xv[c], acc);
      g[gg] = acc;
    }

    const float i0 = sigm(g[0]), i1 = sigm(g[1]);
    const float f0 = sigm(g[2]), f1 = sigm(g[3]);
    const float t0 = tanh_fast(g[4]), t1 = tanh_fast(g[5]);
    const float o0 = sigm(g[6]), o1 = sigm(g[7]);

    c0 = fmaf(f0, c0, i0 * t0);
    c1 = fmaf(f1, c1, i1 * t1);
    h0 = o0 * tanh_fast(c0);
    h1 = o1 * tanh_fast(c1);

    float2 hv = make_float2(h0, h1);
    *(float2*)(out + ((size_t)tt * NB + (size_t)b) * 4 + out_off) = hv;
  }
}

// ---------------------------------------------------------------------------
// FC1: act[b, n] = relu( sum_k relu(h2[b,k]) * w[n,k] + bias[n] ), n<64, k<400
// h2 stored (T, B, 4); k = t*4 + j  -> contiguous 4 floats at h2[(t*NB+b)*4].
// One wave per 16x16 output tile; K stepped by 4 with fp32 WMMA.
// A (16x4 f32): lane%16 = row M, lane/16 picks K-pair {0,1}/{2,3}.
// B (4x16 f32): VGPR v, lane l: K = (l/16)*2 + v, N = l%16.
// D (16x16 f32): VGPR r, lane l: M = r + (l/16)*8, N = l%16.
// ---------------------------------------------------------------------------
__global__ __launch_bounds__(256) void fc1_wmma_kernel(
    const float* __restrict__ h2,    // (T, B, 4)
    const float* __restrict__ w,     // (64, 400)
    const float* __restrict__ bias,  // (64)
    float* __restrict__ act)         // (B, 64)
{
  const int wid = blockIdx.x * (blockDim.x >> 5) + (threadIdx.x >> 5);
  const int lane = threadIdx.x & 31;
  const int m_tile = wid >> 2;        // 4096 tiles of 16 samples
  const int n_tile = wid & 3;         // 4 tiles of 16 outputs
  const int row = lane & 15;
  const int half = lane >> 4;
  const int base_b = m_tile * 16;
  const int n = n_tile * 16 + row;    // output channel for B-load and store

  v8f acc = {};
  const float* wrow = w + (size_t)n * 400 + half * 2;
  const float* arow = h2 + ((size_t)(base_b + row)) * 4 + half * 2;

#pragma unroll 4
  for (int t = 0; t < NT; ++t) {
    v2f a = *(const v2f*)(arow + (size_t)t * NB * 4);
    a.x = fmaxf(a.x, 0.0f);           // relu(h2) before FC1
    a.y = fmaxf(a.y, 0.0f);
    v2f bv = *(const v2f*)(wrow + t * 4);
    acc = __builtin_amdgcn_wmma_f32_16x16x4_f32(
        false, a, false, bv, (short)0, acc, false, false);
  }

  const float bn = bias[n];
#pragma unroll
  for (int r = 0; r < 8; ++r) {
    const int m = r + half * 8;
    act[(size_t)(base_b + m) * 64 + n] = fmaxf(acc[r] + bn, 0.0f);
  }
}

// ---------------------------------------------------------------------------
// FC2: out[b, n] = sum_k act[b,k] * w[n,k] + bias[n], n<20, k<64.
// N covered by 2 tiles of 16 (lanes with n>=20 masked on load/store).
// ---------------------------------------------------------------------------
__global__ __launch_bounds__(256) void fc2_wmma_kernel(
    const float* __restrict__ act,   // (B, 64)
    const float* __restrict__ w,     // (20, 64)
    const float* __restrict__ bias,  // (20)
    float* __restrict__ out)         // (B, 20)
{
  const int wid = blockIdx.x * (blockDim.x >> 5) + (threadIdx.x >> 5);
  const int lane = threadIdx.x & 31;
  const int m_tile = wid >> 1;        // 4096
  const int n_tile = wid & 1;         // 2
  const int row = lane & 15;
  const int half = lane >> 4;
  const int base_b = m_tile * 16;
  const int n = n_tile * 16 + row;
  const bool nvalid = (n < 20);

  v8f acc = {};
  const float* arow = act + (size_t)(base_b + row) * 64 + half * 2;
  const float* wrow = w + (size_t)(nvalid ? n : 0) * 64 + half * 2;

#pragma unroll 4
  for (int kk = 0; kk < 16; ++kk) {
    v2f a = *(const v2f*)(arow + kk * 4);
    v2f bv = *(const v2f*)(wrow + kk * 4);
    if (!nvalid) { bv.x = 0.0f; bv.y = 0.0f; }
    acc = __builtin_amdgcn_wmma_f32_16x16x4_f32(
        false, a, false, bv, (short)0, acc, false, false);
  }

  if (nvalid) {
    const float bn = bias[n];
#pragma unroll
    for (int r = 0; r < 8; ++r) {
      const int m = r + half * 8;
      out[(size_t)(base_b + m) * 20 + n] = acc[r] + bn;
    }
  }
}

// ---------------------------------------------------------------------------
// Host launcher. Input order (setup_inputs): x, then for l0/l0r/l1/l1r:
// (w_ih, w_hh, b_ih, b_hh), then fc1_w, fc1_b, fc2_w, fc2_b.
// Workspace: ws0 = (T,B,4) layer-0 out  : 26,214,400 floats
//            ws1 = (T,B,4) layer-1 out  : 26,214,400 floats
//            ws2 = (B,64) FC1 act, aliases ws0 (layer-0 data dead by then)
// ---------------------------------------------------------------------------
extern "C" void kernel_launch(void* const* d_in, const int* in_sizes, int n_in,
                              void* d_out, int out_size, void* d_ws,
                              size_t ws_size, hipStream_t stream) {
  const float* x = (const float*)d_in[0];
  const float* w_ih_l0  = (const float*)d_in[1];
  const float* w_hh_l0  = (const float*)d_in[2];
  const float* b_ih_l0  = (const float*)d_in[3];
  const float* b_hh_l0  = (const float*)d_in[4];
  const float* w_ih_l0r = (const float*)d_in[5];
  const float* w_hh_l0r = (const float*)d_in[6];
  const float* b_ih_l0r = (const float*)d_in[7];
  const float* b_hh_l0r = (const float*)d_in[8];
  const float* w_ih_l1  = (const float*)d_in[9];
  const float* w_hh_l1  = (const float*)d_in[10];
  const float* b_ih_l1  = (const float*)d_in[11];
  const float* b_hh_l1  = (const float*)d_in[12];
  const float* w_ih_l1r = (const float*)d_in[13];
  const float* w_hh_l1r = (const float*)d_in[14];
  const float* b_ih_l1r = (const float*)d_in[15];
  const float* b_hh_l1r = (const float*)d_in[16];
  const float* fc1_w = (const float*)d_in[17];
  const float* fc1_b = (const float*)d_in[18];
  const float* fc2_w = (const float*)d_in[19];
  const float* fc2_b = (const float*)d_in[20];
  float* out = (float*)d_out;

  float* ws  = (float*)d_ws;
  float* ws0 = ws;                          // (T,B,4) layer-0 output
  float* ws1 = ws + (size_t)NT * NB * 4;    // (T,B,4) layer-1 output
  float* ws2 = ws;                          // (B,64) FC1 act (aliases ws0)

  const dim3 blk(256);
  const dim3 grid_lstm(NB / 256);

  // Layer 0, forward + reverse (write channel pairs 0-1 / 2-3 of ws0)
  lstm_dir_kernel<7, false, false><<<grid_lstm, blk, 0, stream>>>(
      x, w_ih_l0, w_hh_l0, b_ih_l0, b_hh_l0, ws0, 0);
  lstm_dir_kernel<7, true, false><<<grid_lstm, blk, 0, stream>>>(
      x, w_ih_l0r, w_hh_l0r, b_ih_l0r, b_hh_l0r, ws0, 2);

  // Layer 1 over ws0 (T,B,4), forward + reverse -> ws1
  lstm_dir_kernel<4, false, true><<<grid_lstm, blk, 0, stream>>>(
      ws0, w_ih_l1, w_hh_l1, b_ih_l1, b_hh_l1, ws1, 0);
  lstm_dir_kernel<4, true, true><<<grid_lstm, blk, 0, stream>>>(
      ws0, w_ih_l1r, w_hh_l1r, b_ih_l1r, b_hh_l1r, ws1, 2);

  // FC1: (B/16)*(64/16) = 16384 waves -> 2048 blocks of 8 waves
  fc1_wmma_kernel<<<dim3(2048), blk, 0, stream>>>(ws1, fc1_w, fc1_b, ws2);

  // FC2: (B/16)*2 = 8192 waves -> 1024 blocks
  fc2_wmma_kernel<<<dim3(1024), blk, 0, stream>>>(ws2, fc2_w, fc2_b, out);
}